// TGMamba_6021544149832
// MI455X (gfx1250) — compile-verified
//
#include <hip/hip_runtime.h>
#include <hip/hip_bf16.h>
#include <math.h>

typedef float v2f __attribute__((ext_vector_type(2)));
typedef float v8f __attribute__((ext_vector_type(8)));

#define VPB     19          // vertices per graph block
#define NBLK    32
#define NN      608         // total nodes
#define LSEQ    256
#define MROWS   (NN * LSEQ) // 155648 rows for the big GEMMs
#define XDBL_LD 48          // padded x_dbl leading dim: [0:2)=dtr, [4:20)=B, [20:36)=C

// ---------------------------------------------------------------------------
// fp32 WMMA 16x16 tile: C += A(16xK) * B(Kx16), arbitrary strides, K % 4 == 0.
// A-frag layout (ISA 7.12.2): lanes 0-15 hold K={k0,k0+1}, lanes 16-31 K={k0+2,k0+3}.
// C/D layout: VGPR r -> rows {r, r+8} split across lane halves.
// ---------------------------------------------------------------------------
__device__ __forceinline__ v8f wmma_tile(const float* A, int sAr, int sAc,
                                         const float* B, int sBr, int sBc,
                                         int K, v8f acc) {
  const int lane = threadIdx.x & 31;
  const int h    = lane >> 4;
  const int mn   = lane & 15;
  for (int k0 = 0; k0 < K; k0 += 4) {
    const int ka = k0 + 2 * h;
    v2f a, b;
    a.x = A[mn * sAr + ka * sAc];
    a.y = A[mn * sAr + (ka + 1) * sAc];
    b.x = B[ka * sBr + mn * sBc];
    b.y = B[(ka + 1) * sBr + mn * sBc];
    acc = __builtin_amdgcn_wmma_f32_16x16x4_f32(false, a, false, b,
                                                (short)0, acc, false, false);
  }
  return acc;
}

__device__ __forceinline__ void wmma_store(float* C, int ldc, v8f acc) {
  const int lane = threadIdx.x & 31;
  const int h    = lane >> 4;
  const int n    = lane & 15;
#pragma unroll
  for (int r = 0; r < 8; ++r) C[(size_t)(r + 8 * h) * ldc + n] = acc[r];
}

// ---------------------------------------------------------------------------
// CDNA5 async global->LDS copy (ASYNCcnt-tracked), B128 per lane.
// LDS byte address = low 32 bits of the generic pointer (ISA 10.2 aperture rule).
// ---------------------------------------------------------------------------
__device__ __forceinline__ void async_copy_b128(float* lds_dst, const float* gsrc) {
  unsigned lds_off = (unsigned)(unsigned long long)lds_dst;
  asm volatile("global_load_async_to_lds_b128 %0, %1, off"
               :: "v"(lds_off), "v"(gsrc) : "memory");
}
__device__ __forceinline__ void wait_async0() {
  asm volatile("s_wait_asynccnt 0x0" ::: "memory");
}

// ---------------------------------------------------------------------------
// GCN normalization precompute
// ---------------------------------------------------------------------------
__global__ void k_zero(float* deg, float* AhatG) {
  int i = blockIdx.x * blockDim.x + threadIdx.x;
  if (i < NN) deg[i] = 0.f;
  if (i < NBLK * 1024) AhatG[i] = 0.f;
}

__global__ void k_deg(const int* __restrict__ eidx, const float* __restrict__ ew,
                      int E, float* deg) {
  int i = blockIdx.x * blockDim.x + threadIdx.x;
  if (i < E) {
    atomicAdd(&deg[eidx[E + i]], ew[i]);          // dst
  } else if (i < E + NN) {
    atomicAdd(&deg[i - E], 1.0f);                 // self loop weight 1
  }
}

__global__ void k_dinv(const float* deg, float* dinv) {
  int i = blockIdx.x * blockDim.x + threadIdx.x;
  if (i < NN) dinv[i] = rsqrtf(fmaxf(deg[i], 1e-12f));
}

// Dense per-block normalized adjacency: AhatG[b][dst_local(32)][src_local(32)]
__global__ void k_ahat(const int* __restrict__ eidx, const float* __restrict__ ew,
                       int E, const float* __restrict__ dinv, float* AhatG) {
  int i = blockIdx.x * blockDim.x + threadIdx.x;
  if (i < E) {
    int s = eidx[i], d = eidx[E + i];
    int b = d / VPB;
    unsigned ls = (unsigned)(s - b * VPB), ld = (unsigned)(d - b * VPB);
    if (ls < VPB && ld < VPB) {
      atomicAdd(&AhatG[b * 1024 + ld * 32 + ls], ew[i] * dinv[s] * dinv[d]);
    }
  } else if (i < E + NN) {
    int g = i - E, b = g / VPB, lv = g - b * VPB;
    atomicAdd(&AhatG[b * 1024 + lv * 32 + lv], dinv[g] * dinv[g]);
  }
}

// Permuted + zero-padded x_proj_w (48 x 64):
// row 0-1 <- dtr rows 0-1; rows 4..35 <- src rows 2..33 (B then C); rest zero.
__global__ void k_wpad(const float* __restrict__ W, float* Wxp) {
  int i = blockIdx.x * blockDim.x + threadIdx.x;
  if (i < 48 * 64) {
    int r = i >> 6, k = i & 63;
    float v = 0.f;
    if (r < 2) v = W[r * 64 + k];
    else if (r >= 4 && r < 36) v = W[(r - 2) * 64 + k];
    Wxp[i] = v;
  }
}

// ---------------------------------------------------------------------------
// GEMM 1: XZ[M,128] = x_in[M,32] @ in_proj_w^T   (grid 9728, 8 waves = 8 N-tiles)
// ---------------------------------------------------------------------------
__global__ void k_gemm_in(const float* __restrict__ X, const float* __restrict__ W,
                          float* __restrict__ XZ) {
  const int mt = blockIdx.x, wave = threadIdx.x >> 5;
  v8f acc = {0, 0, 0, 0, 0, 0, 0, 0};
  acc = wmma_tile(X + (size_t)mt * 16 * 32, 32, 1,
                  W + wave * 16 * 32, 1, 32, 32, acc);     // B[k][n] = W[n][k]
  wmma_store(XZ + (size_t)mt * 16 * 128 + wave * 16, 128, acc);
}

// GEMM 2: XDBL[M,48] = x[M,64] @ Wxp^T  (x = XZ[:, :64]) (grid 9728, 3 of 4 waves)
__global__ void k_gemm_xdbl(const float* __restrict__ XZ, const float* __restrict__ Wxp,
                            float* __restrict__ XDBL) {
  const int mt = blockIdx.x, wave = threadIdx.x >> 5;
  if (wave < 3) {
    v8f acc = {0, 0, 0, 0, 0, 0, 0, 0};
    acc = wmma_tile(XZ + (size_t)mt * 16 * 128, 128, 1,
                    Wxp + wave * 16 * 64, 1, 64, 64, acc);
    wmma_store(XDBL + (size_t)mt * 16 * XDBL_LD + wave * 16, XDBL_LD, acc);
  }
}

// dt[M,64] = softplus(dtr @ dt_proj_w^T + b)
__global__ void k_dt(const float* __restrict__ XDBL, const float* __restrict__ dtw,
                     const float* __restrict__ dtb, float* __restrict__ DT) {
  size_t idx = (size_t)blockIdx.x * blockDim.x + threadIdx.x;
  if (idx < (size_t)MROWS * 64) {
    size_t m = idx >> 6;
    int d = (int)(idx & 63);
    float v = XDBL[m * XDBL_LD] * dtw[d * 2] + XDBL[m * XDBL_LD + 1] * dtw[d * 2 + 1] + dtb[d];
    DT[idx] = (v > 20.f) ? v : log1pf(__expf(v));
  }
}

// GEMM 3: OUT[M,32] = Y[M,64] @ out_proj_w^T  (grid 9728, 2 waves)
__global__ void k_gemm_out(const float* __restrict__ Y, const float* __restrict__ W,
                           float* __restrict__ OUT) {
  const int mt = blockIdx.x, wave = threadIdx.x >> 5;
  v8f acc = {0, 0, 0, 0, 0, 0, 0, 0};
  acc = wmma_tile(Y + (size_t)mt * 16 * 64, 64, 1,
                  W + wave * 16 * 64, 1, 64, 64, acc);
  wmma_store(OUT + (size_t)mt * 16 * 32 + wave * 16, 32, acc);
}

// ---------------------------------------------------------------------------
// Selective scan: one workgroup per graph block, state LDS-resident.
// Smat[e][v*16+n] (64 x 320, cols 304..319 zero-pad) holds s[v,e,n].
// Per step: Tmat = W x Smat (WMMA), sA = Ahat x Tmat-view (WMMA, fused update).
// Per-step operands double-buffered in LDS, filled by async global->LDS B128
// copies issued one step ahead (overlaps the whole step's compute).
// ---------------------------------------------------------------------------
__global__ void k_scan(const float* __restrict__ XZ, const float* __restrict__ XDBL,
                       const float* __restrict__ DTg,
                       const float* __restrict__ gAw, const float* __restrict__ gAb,
                       const float* __restrict__ gBw, const float* __restrict__ gBb,
                       const float* __restrict__ gCw, const float* __restrict__ gCb,
                       const float* __restrict__ Alog, const float* __restrict__ Dp,
                       const float* __restrict__ AhatG, float* __restrict__ Yg) {
  extern __shared__ float lds[];
  float* Smat  = lds;               // 64*320 = 20480
  float* Tmat  = Smat + 20480;      // 20480
  float* wA    = Tmat + 20480;      // 4096
  float* wB    = wA + 4096;         // 4096
  float* wC    = wB + 4096;         // 4096
  float* AhatS = wC + 4096;         // 32*32
  float* Aneg  = AhatS + 1024;      // 64*16
  float* dtBuf = Aneg + 1024;       // 2 x 1216 (double buffered)
  float* uBuf  = dtBuf + 2432;      // 2 x 1216
  float* zBuf  = uBuf + 2432;       // 2 x 1216
  float* uBs   = zBuf + 2432;       // 1216
  float* TBs   = uBs + 1216;        // 1216
  float* ysS   = TBs + 1216;        // 1216
  float* BBuf  = ysS + 1216;        // 2 x 304
  float* CBuf  = BBuf + 608;        // 2 x 304   (total 67456 floats = 264 KB)

  const int tid  = threadIdx.x;
  const int wave = tid >> 5;
  const int lane = tid & 31;
  const int h    = lane >> 4;
  const int n    = lane & 15;
  const int b    = blockIdx.x;
  const int g0   = b * VPB;

  // ---- async prefetch of step-t operands into slot p (B128 per lane) ----
  auto prefetch = [&](int t, int p) {
    for (int c = tid; c < 1064; c += 256) {
      if (c < 912) {                         // dt / u / z : 304 chunks each
        int kind = c / 304;
        int cc = c - kind * 304;
        int v = cc >> 4, d4 = (cc & 15) * 4;
        size_t m = (size_t)(g0 + v) * LSEQ + t;
        if (kind == 0)      async_copy_b128(dtBuf + p * 1216 + cc * 4, DTg + m * 64 + d4);
        else if (kind == 1) async_copy_b128(uBuf  + p * 1216 + cc * 4, XZ + m * 128 + d4);
        else                async_copy_b128(zBuf  + p * 1216 + cc * 4, XZ + m * 128 + 64 + d4);
      } else {                               // B / C : 76 chunks each
        int cc = c - 912;
        int kind = cc / 76;
        int c2 = cc - kind * 76;
        int v = c2 >> 2, n4 = (c2 & 3) * 4;
        size_t m = (size_t)(g0 + v) * LSEQ + t;
        const float* g = XDBL + m * XDBL_LD + (kind ? 20 : 4) + n4;
        async_copy_b128((kind ? CBuf : BBuf) + p * 304 + c2 * 4, g);
      }
    }
  };

  prefetch(0, 0);                                                 // overlap with init

  for (int i = tid; i < 40960; i += 256) Smat[i] = 0.f;           // Smat + Tmat
  for (int i = tid; i < 4096; i += 256) { wA[i] = gAw[i]; wB[i] = gBw[i]; wC[i] = gCw[i]; }
  for (int i = tid; i < 1024; i += 256) {
    AhatS[i] = AhatG[b * 1024 + i];
    Aneg[i]  = -__expf(Alog[i]);                                  // A = -exp(A_log)
  }

  for (int t = 0; t < LSEQ; ++t) {
    const int p = t & 1;
    float* dtS = dtBuf + p * 1216;
    float* uS  = uBuf + p * 1216;
    float* zS  = zBuf + p * 1216;
    float* BS  = BBuf + p * 304;
    float* CS  = CBuf + p * 304;

    wait_async0();          // own async LDS writes complete...
    __syncthreads();        // ...then visible to all waves

    if (t + 1 < LSEQ) prefetch(t + 1, p ^ 1);   // overlap next-step copy

    // -------- gB path stage 1: TB[u][d] = sum_e gB_w[d][e] * u[u][e] --------
    for (int i = tid; i < 1216; i += 256) {
      int u = i >> 6, d = i & 63;
      float s = 0.f;
#pragma unroll 8
      for (int e = 0; e < 64; ++e) s += wB[d * 64 + e] * uS[u * 64 + e];
      TBs[i] = s;
    }
    __syncthreads();

    // -------- gB path stage 2 (aggregation) + gA transform (WMMA) --------
    for (int i = tid; i < 1216; i += 256) {
      int v = i >> 6, d = i & 63;
      float s = gBb[d];
      for (int u = 0; u < VPB; ++u) s += AhatS[v * 32 + u] * TBs[u * 64 + d];
      uBs[i] = s;
    }
    for (int tile = wave; tile < 76; tile += 8) {                 // 4 x 19 tiles, K=64
      int rd = tile / 19, ct = tile % 19;
      v8f acc = {0, 0, 0, 0, 0, 0, 0, 0};
      acc = wmma_tile(wA + rd * 16 * 64, 64, 1, Smat + ct * 16, 320, 1, 64, acc);
      wmma_store(Tmat + rd * 16 * 320 + ct * 16, 320, acc);
    }
    __syncthreads();

    // -------- sA = Ahat x Tmat-view (WMMA), fused state update -> Smat --------
    for (int tile = wave; tile < 128; tile += 8) {                // 2 v-tiles x 64 d
      int vt = tile & 1, dcol = tile >> 1;
      v8f acc = {0, 0, 0, 0, 0, 0, 0, 0};
      acc = wmma_tile(AhatS + vt * 16 * 32, 32, 1, Tmat + dcol * 320, 16, 1, 20, acc);
#pragma unroll
      for (int r = 0; r < 8; ++r) {
        int v = vt * 16 + r + 8 * h;
        if (v < VPB) {
          float dtv = dtS[v * 64 + dcol];
          float dA  = __expf(dtv * Aneg[dcol * 16 + n]);
          float sAv = acc[r] + gAb[dcol];
          Smat[dcol * 320 + v * 16 + n] = sAv * dA + uBs[v * 64 + dcol] * dtv * BS[v * 16 + n];
        }
      }
    }
    __syncthreads();

    // -------- gC transform (WMMA) --------
    for (int tile = wave; tile < 76; tile += 8) {
      int rd = tile / 19, ct = tile % 19;
      v8f acc = {0, 0, 0, 0, 0, 0, 0, 0};
      acc = wmma_tile(wC + rd * 16 * 64, 64, 1, Smat + ct * 16, 320, 1, 64, acc);
      wmma_store(Tmat + rd * 16 * 320 + ct * 16, 320, acc);
    }
    __syncthreads();

    // -------- sC aggregation (WMMA) + y = sC . C  (cross-lane reduce) --------
    for (int tile = wave; tile < 128; tile += 8) {
      int vt = tile & 1, dcol = tile >> 1;
      v8f acc = {0, 0, 0, 0, 0, 0, 0, 0};
      acc = wmma_tile(AhatS + vt * 16 * 32, 32, 1, Tmat + dcol * 320, 16, 1, 20, acc);
#pragma unroll
      for (int r = 0; r < 8; ++r) {
        int v = vt * 16 + r + 8 * h;
        float val = (v < VPB) ? (acc[r] + gCb[dcol]) * CS[v * 16 + n] : 0.f;
        val += __shfl_xor(val, 1, 32);
        val += __shfl_xor(val, 2, 32);
        val += __shfl_xor(val, 4, 32);
        val += __shfl_xor(val, 8, 32);
        if ((n == 0) && v < VPB) ysS[v * 64 + dcol] = val;
      }
    }
    __syncthreads();

    // -------- finalize: y = (sC.C + D*u) * silu(z) --------
    for (int i = tid; i < 1216; i += 256) {
      int v = i >> 6, d = i & 63;
      float zv = zS[i];
      float y = ysS[i] + Dp[d] * uS[i];
      y *= zv * (1.f / (1.f + __expf(-zv)));
      Yg[((size_t)(g0 + v) * LSEQ + t) * 64 + d] = y;
    }
    __syncthreads();
  }
}

// ---------------------------------------------------------------------------
extern "C" void kernel_launch(void* const* d_in, const int* in_sizes, int n_in,
                              void* d_out, int out_size, void* d_ws, size_t ws_size,
                              hipStream_t stream) {
  const float* x_in      = (const float*)d_in[0];
  const int*   eidx      = (const int*)d_in[1];
  const float* ew        = (const float*)d_in[2];
  const float* in_proj_w = (const float*)d_in[3];
  const float* x_proj_w  = (const float*)d_in[4];
  const float* dt_proj_w = (const float*)d_in[5];
  const float* dt_proj_b = (const float*)d_in[6];
  const float* A_log     = (const float*)d_in[7];
  const float* D_param   = (const float*)d_in[8];
  const float* out_proj_w= (const float*)d_in[9];
  const float* gA_w      = (const float*)d_in[10];
  const float* gA_b      = (const float*)d_in[11];
  const float* gB_w      = (const float*)d_in[12];
  const float* gB_b      = (const float*)d_in[13];
  const float* gC_w      = (const float*)d_in[14];
  const float* gC_b      = (const float*)d_in[15];
  const int E = in_sizes[2];

  float* ws    = (float*)d_ws;
  float* XZ    = ws;                               // MROWS*128
  float* XDBL  = XZ + (size_t)MROWS * 128;         // MROWS*48
  float* DT    = XDBL + (size_t)MROWS * XDBL_LD;   // MROWS*64
  float* Y     = DT + (size_t)MROWS * 64;          // MROWS*64
  float* deg   = Y + (size_t)MROWS * 64;           // NN
  float* dinv  = deg + NN;                         // NN
  float* AhatG = dinv + NN;                        // NBLK*1024
  float* Wxp   = AhatG + (size_t)NBLK * 1024;      // 48*64

  // GCN normalization precompute
  k_zero<<<(NBLK * 1024 + 255) / 256, 256, 0, stream>>>(deg, AhatG);
  k_deg <<<(E + NN + 255) / 256, 256, 0, stream>>>(eidx, ew, E, deg);
  k_dinv<<<(NN + 255) / 256, 256, 0, stream>>>(deg, dinv);
  k_ahat<<<(E + NN + 255) / 256, 256, 0, stream>>>(eidx, ew, E, dinv, AhatG);
  k_wpad<<<(48 * 64 + 255) / 256, 256, 0, stream>>>(x_proj_w, Wxp);

  // Projections (fp32 WMMA GEMMs)
  k_gemm_in  <<<MROWS / 16, 256, 0, stream>>>(x_in, in_proj_w, XZ);
  k_gemm_xdbl<<<MROWS / 16, 128, 0, stream>>>(XZ, Wxp, XDBL);
  {
    size_t total = (size_t)MROWS * 64;
    k_dt<<<(unsigned)((total + 255) / 256), 256, 0, stream>>>(XDBL, dt_proj_w, dt_proj_b, DT);
  }

  // Selective scan: 32 workgroups (one per graph block), 264 KB dynamic LDS each
  const size_t scan_lds = 67456 * sizeof(float);
  k_scan<<<NBLK, 256, scan_lds, stream>>>(XZ, XDBL, DT, gA_w, gA_b, gB_w, gB_b,
                                          gC_w, gC_b, A_log, D_param, AhatG, Y);

  // Output projection
  k_gemm_out<<<MROWS / 16, 64, 0, stream>>>(Y, out_proj_w, (float*)d_out);
}